// DifferentiableRobotModel_19533511262454
// MI455X (gfx1250) — compile-verified
//
#include <hip/hip_runtime.h>
#include <hip/hip_bf16.h>
#include <stdint.h>

#define N_LINKS 12
#define TPB 128                 // threads per block = 4 wave32
#define ROW_F 144               // floats of output per batch element
#define ROW_F4 36               // float4s per batch element
#define ROW_PAD_F 148           // padded LDS row stride (floats), 16B aligned
#define ROW_PAD_F4 37
#define WS_STRIDE 32            // floats per link in workspace (30 used, 128B aligned)

// ---------------------------------------------------------------------------
// Precompute per-link constants:  Rf @ Rodrigues(a, th) = c*C + s*B + P
//   B = Rf @ skew(a),  u = Rf @ a,  P = u a^T,  C = Rf - P
// ws[link*32 + 0..8]=C, [9..17]=B, [18..26]=P, [27..29]=tf
// ---------------------------------------------------------------------------
__global__ void fk_precompute_kernel(const float* __restrict__ axes,
                                     const float* __restrict__ rot_fixed,
                                     const float* __restrict__ trans_fixed,
                                     float* __restrict__ ws)
{
    const int i = threadIdx.x;
    if (i >= N_LINKS) return;

    const float x = axes[3*i+0], y = axes[3*i+1], z = axes[3*i+2];
    float f[9];
    #pragma unroll
    for (int k = 0; k < 9; ++k) f[k] = rot_fixed[9*i+k];

    float* w = ws + i * WS_STRIDE;
    #pragma unroll
    for (int r = 0; r < 3; ++r) {
        const float fr0 = f[3*r+0], fr1 = f[3*r+1], fr2 = f[3*r+2];
        // B row r = Rf row r @ K(a)
        const float b0 = fr1*z - fr2*y;
        const float b1 = fr2*x - fr0*z;
        const float b2 = fr0*y - fr1*x;
        // u_r = Rf row r . a ; P row r = u_r * a ; C = Rf - P
        const float u  = fr0*x + fr1*y + fr2*z;
        const float p0 = u*x, p1 = u*y, p2 = u*z;
        w[3*r+0]      = fr0 - p0;   // C
        w[3*r+1]      = fr1 - p1;
        w[3*r+2]      = fr2 - p2;
        w[9  + 3*r+0] = b0;         // B
        w[9  + 3*r+1] = b1;
        w[9  + 3*r+2] = b2;
        w[18 + 3*r+0] = p0;         // P
        w[18 + 3*r+1] = p1;
        w[18 + 3*r+2] = p2;
    }
    w[27] = trans_fixed[3*i+0];
    w[28] = trans_fixed[3*i+1];
    w[29] = trans_fixed[3*i+2];
    w[30] = 0.f; w[31] = 0.f;
}

// ---------------------------------------------------------------------------
// Shared tail: stage pose row into LDS, then async LDS->global b128 stream.
// ---------------------------------------------------------------------------
__device__ __forceinline__ void fk_store_tile(float* tile, int tid, int blockId,
                                              float* out, int B)
{
    __syncthreads();   // all ds_store complete (dscnt) before async engine reads LDS

    const size_t base4 = (size_t)blockId * (TPB * ROW_F4);
    float4* out4 = reinterpret_cast<float4*>(out);

    const bool full = (size_t)(blockId + 1) * TPB <= (size_t)B;  // uniform
    if (full) {
        #pragma unroll
        for (int k = 0; k < ROW_F4; ++k) {
            const int g  = k * TPB + tid;
            const int bl = g / ROW_F4;
            const int c4 = g - bl * ROW_F4;
            // LDS operand derived from &tile[...] so the shared array escapes
            const unsigned lds_byte =
                (unsigned)(uintptr_t)(tile + (size_t)(bl * ROW_PAD_F4 + c4) * 4);
            const unsigned long long gaddr =
                (unsigned long long)(uintptr_t)(out4 + base4 + (size_t)g);
            asm volatile("global_store_async_from_lds_b128 %0, %1, off"
                         :: "v"(gaddr), "v"(lds_byte)
                         : "memory");
        }
    } else {
        const size_t total4 = (size_t)B * ROW_F4;
        #pragma unroll
        for (int k = 0; k < ROW_F4; ++k) {
            const int g  = k * TPB + tid;
            const size_t g4 = base4 + (size_t)g;
            if (g4 < total4) {
                const int bl = g / ROW_F4;
                const int c4 = g - bl * ROW_F4;
                const unsigned lds_byte =
                    (unsigned)(uintptr_t)(tile + (size_t)(bl * ROW_PAD_F4 + c4) * 4);
                const unsigned long long gaddr =
                    (unsigned long long)(uintptr_t)(out4 + g4);
                asm volatile("global_store_async_from_lds_b128 %0, %1, off"
                             :: "v"(gaddr), "v"(lds_byte)
                             : "memory");
            }
        }
    }
    asm volatile("s_wait_asynccnt 0x0" ::: "memory");
}

// ---------------------------------------------------------------------------
// Fast path: per-link joint transform is 2 FMAs/entry from precomputed C,B,P.
// ---------------------------------------------------------------------------
__global__ __launch_bounds__(TPB) void fk_chain_fast(
    const float* __restrict__ q,    // [B,12]
    const float* __restrict__ ws,   // [12,32] precomputed C,B,P,tf (uniform -> s_load)
    float* __restrict__ out,        // [B,12,12]
    int B)
{
    __shared__ float tile[TPB * ROW_PAD_F];   // 75,776 B

    const int tid = threadIdx.x;
    const int b   = blockIdx.x * TPB + tid;
    float* myrow  = &tile[tid * ROW_PAD_F];

    if (b < B) {
        float qv[N_LINKS];
        const float4* q4 = reinterpret_cast<const float4*>(q + (size_t)b * N_LINKS);
        #pragma unroll
        for (int k = 0; k < 3; ++k) {
            float4 v = q4[k];
            qv[4*k+0] = v.x; qv[4*k+1] = v.y; qv[4*k+2] = v.z; qv[4*k+3] = v.w;
        }

        float r00=1.f, r01=0.f, r02=0.f;
        float r10=0.f, r11=1.f, r12=0.f;
        float r20=0.f, r21=0.f, r22=1.f;
        float t0=0.f, t1=0.f, t2=0.f;

        #pragma unroll
        for (int i = 0; i < N_LINKS; ++i) {
            const float* w = ws + i * WS_STRIDE;

            float s, c;
            __sincosf(qv[i], &s, &c);

            // Rj = c*C + s*B + P   (2 FMAs per entry)
            const float j00 = fmaf(c, w[0], fmaf(s, w[ 9], w[18]));
            const float j01 = fmaf(c, w[1], fmaf(s, w[10], w[19]));
            const float j02 = fmaf(c, w[2], fmaf(s, w[11], w[20]));
            const float j10 = fmaf(c, w[3], fmaf(s, w[12], w[21]));
            const float j11 = fmaf(c, w[4], fmaf(s, w[13], w[22]));
            const float j12 = fmaf(c, w[5], fmaf(s, w[14], w[23]));
            const float j20 = fmaf(c, w[6], fmaf(s, w[15], w[24]));
            const float j21 = fmaf(c, w[7], fmaf(s, w[16], w[25]));
            const float j22 = fmaf(c, w[8], fmaf(s, w[17], w[26]));

            const float tf0 = w[27], tf1 = w[28], tf2 = w[29];

            // t_new = t + R_old @ tf
            const float nt0 = t0 + fmaf(r00,tf0, fmaf(r01,tf1, r02*tf2));
            const float nt1 = t1 + fmaf(r10,tf0, fmaf(r11,tf1, r12*tf2));
            const float nt2 = t2 + fmaf(r20,tf0, fmaf(r21,tf1, r22*tf2));

            // R_new = R_old @ Rj
            const float n00 = fmaf(r00,j00, fmaf(r01,j10, r02*j20));
            const float n01 = fmaf(r00,j01, fmaf(r01,j11, r02*j21));
            const float n02 = fmaf(r00,j02, fmaf(r01,j12, r02*j22));
            const float n10 = fmaf(r10,j00, fmaf(r11,j10, r12*j20));
            const float n11 = fmaf(r10,j01, fmaf(r11,j11, r12*j21));
            const float n12 = fmaf(r10,j02, fmaf(r11,j12, r12*j22));
            const float n20 = fmaf(r20,j00, fmaf(r21,j10, r22*j20));
            const float n21 = fmaf(r20,j01, fmaf(r21,j11, r22*j21));
            const float n22 = fmaf(r20,j02, fmaf(r21,j12, r22*j22));

            r00=n00; r01=n01; r02=n02;
            r10=n10; r11=n11; r12=n12;
            r20=n20; r21=n21; r22=n22;
            t0=nt0;  t1=nt1;  t2=nt2;

            float4* p = reinterpret_cast<float4*>(myrow + i * 12);
            p[0] = make_float4(r00, r01, r02, r10);
            p[1] = make_float4(r11, r12, r20, r21);
            p[2] = make_float4(r22, t0,  t1,  t2);
        }
    }

    fk_store_tile(tile, tid, blockIdx.x, out, B);
}

// ---------------------------------------------------------------------------
// Self-contained fallback (if d_ws is too small): computes constants inline.
// ---------------------------------------------------------------------------
__global__ __launch_bounds__(TPB) void fk_chain_fallback(
    const float* __restrict__ q,
    const float* __restrict__ axes,
    const float* __restrict__ rot_fixed,
    const float* __restrict__ trans_fixed,
    float* __restrict__ out,
    int B)
{
    __shared__ float tile[TPB * ROW_PAD_F];

    const int tid = threadIdx.x;
    const int b   = blockIdx.x * TPB + tid;
    float* myrow  = &tile[tid * ROW_PAD_F];

    if (b < B) {
        float qv[N_LINKS];
        const float4* q4 = reinterpret_cast<const float4*>(q + (size_t)b * N_LINKS);
        #pragma unroll
        for (int k = 0; k < 3; ++k) {
            float4 v = q4[k];
            qv[4*k+0] = v.x; qv[4*k+1] = v.y; qv[4*k+2] = v.z; qv[4*k+3] = v.w;
        }

        float r00=1.f, r01=0.f, r02=0.f;
        float r10=0.f, r11=1.f, r12=0.f;
        float r20=0.f, r21=0.f, r22=1.f;
        float t0=0.f, t1=0.f, t2=0.f;

        #pragma unroll
        for (int i = 0; i < N_LINKS; ++i) {
            const float ax = axes[3*i+0], ay = axes[3*i+1], az = axes[3*i+2];
            const float f00 = rot_fixed[9*i+0], f01 = rot_fixed[9*i+1], f02 = rot_fixed[9*i+2];
            const float f10 = rot_fixed[9*i+3], f11 = rot_fixed[9*i+4], f12 = rot_fixed[9*i+5];
            const float f20 = rot_fixed[9*i+6], f21 = rot_fixed[9*i+7], f22 = rot_fixed[9*i+8];
            const float tf0 = trans_fixed[3*i+0], tf1 = trans_fixed[3*i+1], tf2 = trans_fixed[3*i+2];

            float s, c;
            __sincosf(qv[i], &s, &c);
            const float oc = 1.0f - c;

            const float a00 = fmaf(ax*ax, oc,  c);
            const float a01 = fmaf(ax*ay, oc, -az*s);
            const float a02 = fmaf(ax*az, oc,  ay*s);
            const float a10 = fmaf(ay*ax, oc,  az*s);
            const float a11 = fmaf(ay*ay, oc,  c);
            const float a12 = fmaf(ay*az, oc, -ax*s);
            const float a20 = fmaf(az*ax, oc, -ay*s);
            const float a21 = fmaf(az*ay, oc,  ax*s);
            const float a22 = fmaf(az*az, oc,  c);

            const float j00 = fmaf(f00,a00, fmaf(f01,a10, f02*a20));
            const float j01 = fmaf(f00,a01, fmaf(f01,a11, f02*a21));
            const float j02 = fmaf(f00,a02, fmaf(f01,a12, f02*a22));
            const float j10 = fmaf(f10,a00, fmaf(f11,a10, f12*a20));
            const float j11 = fmaf(f10,a01, fmaf(f11,a11, f12*a21));
            const float j12 = fmaf(f10,a02, fmaf(f11,a12, f12*a22));
            const float j20 = fmaf(f20,a00, fmaf(f21,a10, f22*a20));
            const float j21 = fmaf(f20,a01, fmaf(f21,a11, f22*a21));
            const float j22 = fmaf(f20,a02, fmaf(f21,a12, f22*a22));

            const float nt0 = t0 + fmaf(r00,tf0, fmaf(r01,tf1, r02*tf2));
            const float nt1 = t1 + fmaf(r10,tf0, fmaf(r11,tf1, r12*tf2));
            const float nt2 = t2 + fmaf(r20,tf0, fmaf(r21,tf1, r22*tf2));

            const float n00 = fmaf(r00,j00, fmaf(r01,j10, r02*j20));
            const float n01 = fmaf(r00,j01, fmaf(r01,j11, r02*j21));
            const float n02 = fmaf(r00,j02, fmaf(r01,j12, r02*j22));
            const float n10 = fmaf(r10,j00, fmaf(r11,j10, r12*j20));
            const float n11 = fmaf(r10,j01, fmaf(r11,j11, r12*j21));
            const float n12 = fmaf(r10,j02, fmaf(r11,j12, r12*j22));
            const float n20 = fmaf(r20,j00, fmaf(r21,j10, r22*j20));
            const float n21 = fmaf(r20,j01, fmaf(r21,j11, r22*j21));
            const float n22 = fmaf(r20,j02, fmaf(r21,j12, r22*j22));

            r00=n00; r01=n01; r02=n02;
            r10=n10; r11=n11; r12=n12;
            r20=n20; r21=n21; r22=n22;
            t0=nt0;  t1=nt1;  t2=nt2;

            float4* p = reinterpret_cast<float4*>(myrow + i * 12);
            p[0] = make_float4(r00, r01, r02, r10);
            p[1] = make_float4(r11, r12, r20, r21);
            p[2] = make_float4(r22, t0,  t1,  t2);
        }
    }

    fk_store_tile(tile, tid, blockIdx.x, out, B);
}

extern "C" void kernel_launch(void* const* d_in, const int* in_sizes, int n_in,
                              void* d_out, int out_size, void* d_ws, size_t ws_size,
                              hipStream_t stream) {
    const float* q  = (const float*)d_in[0];   // [B,12]
    const float* ax = (const float*)d_in[1];   // [12,3]
    const float* rf = (const float*)d_in[2];   // [12,3,3]
    const float* tf = (const float*)d_in[3];   // [12,3]
    float* out = (float*)d_out;                // [B,12,12]

    const int B = in_sizes[0] / N_LINKS;
    const int blocks = (B + TPB - 1) / TPB;
    const size_t ws_needed = (size_t)N_LINKS * WS_STRIDE * sizeof(float);

    if (d_ws != nullptr && ws_size >= ws_needed) {
        hipLaunchKernelGGL(fk_precompute_kernel, dim3(1), dim3(32), 0, stream,
                           ax, rf, tf, (float*)d_ws);
        hipLaunchKernelGGL(fk_chain_fast, dim3(blocks), dim3(TPB), 0, stream,
                           q, (const float*)d_ws, out, B);
    } else {
        hipLaunchKernelGGL(fk_chain_fallback, dim3(blocks), dim3(TPB), 0, stream,
                           q, ax, rf, tf, out, B);
    }
}